// DipolePredictorE3NN_11450382812015
// MI455X (gfx1250) — compile-verified
//
#include <hip/hip_runtime.h>
#include <hip/hip_bf16.h>

// ---------------------------------------------------------------------------
// DipolePredictorE3NN for MI455X (gfx1250, wave32)
// Phase 1: HBM-bound streaming reduction (176 MiB @ 23.3 TB/s ~ 7.6us floor)
// Phase 2: tiny MLP done with V_WMMA_F32_16X16X4_F32 (full f32 precision)
// ---------------------------------------------------------------------------

#define C01 0.40824829046386301636f   // sqrt(0.5)/sqrt(3)
#define C11 0.28867513459481288225f   // sqrt(0.5)/sqrt(6)

#define BATCH 1024
#define NPTS  4096

typedef __attribute__((ext_vector_type(2))) float v2f;
typedef __attribute__((ext_vector_type(8))) float v8f;

// ---------------------------------------------------------------------------
// Phase 1: per-batch reduction  g[b][0:3] = mean_n tp_out[b][n][0:3]
// One block per batch row; threads stride over n.  feats row is contiguous
// (8 floats/pt -> two b128 loads); edge is stride-3 dword-coalesced.
// ---------------------------------------------------------------------------
__global__ __launch_bounds__(256) void dipole_reduce(
    const float* __restrict__ feats, const float* __restrict__ edge,
    const float* __restrict__ w0p, const float* __restrict__ w1p,
    float* __restrict__ g)
{
    const int b = blockIdx.x;
    const int t = threadIdx.x;

    const float w00 = w0p[0], w01 = w0p[1], w02 = w0p[2], w03 = w0p[3], w04 = w0p[4];
    const float cw1 = C11 * w1p[0];

    const float4* __restrict__ f4 = (const float4*)(feats + (size_t)b * NPTS * 8);
    const float*  __restrict__ eb = edge + (size_t)b * NPTS * 3;

    float ax = 0.f, ay = 0.f, az = 0.f;
    for (int n = t; n < NPTS; n += 256) {
        float4 fa = f4[2 * n];
        float4 fb = f4[2 * n + 1];
        float s  = fa.x * w00 + fa.y * w01 + fa.z * w02 + fa.w * w03 + fb.x * w04;
        float vx = fb.y, vy = fb.z, vz = fb.w;
        float e0 = eb[3 * n], e1 = eb[3 * n + 1], e2 = eb[3 * n + 2];
        float cs = C01 * s;
        ax += cs * e0 + cw1 * (vy * e2 - vz * e1);
        ay += cs * e1 + cw1 * (vz * e0 - vx * e2);
        az += cs * e2 + cw1 * (vx * e1 - vy * e0);
    }

    // wave32 reduction
    #pragma unroll
    for (int off = 16; off > 0; off >>= 1) {
        ax += __shfl_down(ax, off, 32);
        ay += __shfl_down(ay, off, 32);
        az += __shfl_down(az, off, 32);
    }

    __shared__ float red[8][3];
    const int wid = t >> 5;
    if ((t & 31) == 0) { red[wid][0] = ax; red[wid][1] = ay; red[wid][2] = az; }
    __syncthreads();
    if (t < 3) {
        float sum = 0.f;
        #pragma unroll
        for (int w = 0; w < 8; ++w) sum += red[w][t];
        g[b * 3 + t] = sum * (1.0f / (float)NPTS);
    }
}

// ---------------------------------------------------------------------------
// Phase 2: out = relu(g @ W1 + b1) @ W2 + b2 via V_WMMA_F32_16X16X4_F32.
// One wave (32 threads) per 16-row tile of g; 64 tiles.
//
// f32 WMMA operand layouts (ISA 7.12.2):
//   A 16x4 : lane&15 = M,  K = 2*(lane>>4) + vgpr
//   B 4x16 : lane&15 = N,  K = 2*(lane>>4) + vgpr
//   D 16x16: vgpr r, lane -> M = r + 8*(lane>>4), N = lane&15
// ---------------------------------------------------------------------------
#define HPAD 132   // LDS row stride (floats), padded to avoid bank conflicts

__global__ __launch_bounds__(32) void dipole_mlp(
    const float* __restrict__ g,
    const float* __restrict__ W1, const float* __restrict__ b1,
    const float* __restrict__ W2, const float* __restrict__ b2,
    float* __restrict__ out)
{
    const int tile = blockIdx.x;       // 64 tiles of 16 rows
    const int l    = threadIdx.x;      // 0..31, full wave active (EXEC all 1s)
    const int ln   = l & 15;
    const int hf   = l >> 4;
    const int kb   = 2 * hf;           // K base for this lane-half: 0 or 2
    const int row0 = tile * 16;

    __shared__ float hsh[16 * HPAD];

    // ---- GEMM1: h = relu(g_tile(16x4pad) @ W1(4pad x 128) + b1) ----
    v2f a;
    a.x = g[(row0 + ln) * 3 + kb];                       // K = kb (0 or 2)
    a.y = (hf == 0) ? g[(row0 + ln) * 3 + 1] : 0.0f;     // K = kb+1 (1, or 3->pad 0)

    #pragma unroll
    for (int c = 0; c < 8; ++c) {                        // 8 x 16-col chunks of W1
        const int col = c * 16 + ln;
        v2f bv;
        bv.x = W1[kb * 128 + col];                       // K = kb
        bv.y = (hf == 0) ? W1[128 + col] : 0.0f;         // K = kb+1
        v8f acc = {};
        acc = __builtin_amdgcn_wmma_f32_16x16x4_f32(
            false, a, false, bv, (short)0, acc, false, false);
        const float bias = b1[col];
        #pragma unroll
        for (int r = 0; r < 8; ++r) {
            float hv = acc[r] + bias;                    // D: M = r + 8*hf, N = col
            hv = hv > 0.0f ? hv : 0.0f;
            hsh[(r + 8 * hf) * HPAD + col] = hv;
        }
    }
    __syncthreads();

    // ---- GEMM2: out_tile = h(16x128) @ W2(128 x 3pad16) + b2 ----
    v8f acc2 = {};
    #pragma unroll 4
    for (int kk = 0; kk < 32; ++kk) {
        const int k0 = 4 * kk + kb;
        v2f a2;
        a2.x = hsh[ln * HPAD + k0];                      // A: M = ln, K = k0
        a2.y = hsh[ln * HPAD + k0 + 1];                  //    K = k0+1
        v2f bw;
        bw.x = (ln < 3) ? W2[k0 * 3 + ln] : 0.0f;        // B: N = ln (cols >=3 padded)
        bw.y = (ln < 3) ? W2[(k0 + 1) * 3 + ln] : 0.0f;
        acc2 = __builtin_amdgcn_wmma_f32_16x16x4_f32(
            false, a2, false, bw, (short)0, acc2, false, false);
    }

    if (ln < 3) {
        const float bias = b2[ln];
        #pragma unroll
        for (int r = 0; r < 8; ++r)
            out[(row0 + r + 8 * hf) * 3 + ln] = acc2[r] + bias;
    }
}

// ---------------------------------------------------------------------------
extern "C" void kernel_launch(void* const* d_in, const int* in_sizes, int n_in,
                              void* d_out, int out_size, void* d_ws, size_t ws_size,
                              hipStream_t stream) {
    const float* feats = (const float*)d_in[0];   // (1024,4096,8)
    const float* edge  = (const float*)d_in[1];   // (1024,4096,3)
    const float* w0    = (const float*)d_in[2];   // (5,)
    const float* w1    = (const float*)d_in[3];   // (1,)
    const float* W1    = (const float*)d_in[4];   // (3,128)
    const float* b1    = (const float*)d_in[5];   // (128,)
    const float* W2    = (const float*)d_in[6];   // (128,3)
    const float* b2    = (const float*)d_in[7];   // (3,)

    float* g = (float*)d_ws;                      // (1024,3) scratch

    dipole_reduce<<<BATCH, 256, 0, stream>>>(feats, edge, w0, w1, g);
    dipole_mlp<<<BATCH / 16, 32, 0, stream>>>(g, W1, b1, W2, b2, (float*)d_out);
}